// EqxEquivariantMessagePasser_90993177133600
// MI455X (gfx1250) — compile-verified
//
#include <hip/hip_runtime.h>
#include <math.h>

// ---------------------------------------------------------------------------
// EqxEquivariantMessagePasser for MI455X (gfx1250, wave32, WMMA)
// fp32 end-to-end; dense channel GEMMs on V_WMMA_F32_16X16X4_F32.
// ---------------------------------------------------------------------------

typedef float v2f __attribute__((ext_vector_type(2)));
typedef float v8f __attribute__((ext_vector_type(8)));

__device__ __forceinline__ v8f wmma_f32_16x16x4(v2f a, v2f b, v8f c) {
  // 8 args: (neg_a, A, neg_b, B, c_mod, C, reuse_a, reuse_b)
  return __builtin_amdgcn_wmma_f32_16x16x4_f32(false, a, false, b, (short)0, c,
                                               false, false);
}

// ---------------------------------------------------------------------------
// Kernel 0: zero workspace region (pooled accumulators)
// ---------------------------------------------------------------------------
__global__ void zero_kernel(float* __restrict__ p, size_t n) {
  size_t i = (size_t)blockIdx.x * blockDim.x + threadIdx.x;
  if (i < n) p[i] = 0.0f;
}

// ---------------------------------------------------------------------------
// Kernel 1: per-(atom, channel) RMS scale: s = gamma[c] / sqrt(mean_ij f^2 + eps)
// ---------------------------------------------------------------------------
__global__ void rms_scale_kernel(const float* __restrict__ f,
                                 const float* __restrict__ gamma_l,
                                 float* __restrict__ s, int natoms, int sq) {
  int idx = blockIdx.x * blockDim.x + threadIdx.x;
  if (idx >= natoms * 32) return;
  int c = idx & 31;
  int n = idx >> 5;
  const float* fp = f + (size_t)n * sq * 32 + c;
  float acc = 0.0f;
  for (int t = 0; t < sq; ++t) {
    float v = fp[(size_t)t * 32];
    acc += v * v;
  }
  float rms = sqrtf(acc / (float)sq + 1e-6f);
  s[idx] = gamma_l[c] / rms;
}

// ---------------------------------------------------------------------------
// Kernel 2: h[r, g] = sum_k (f[r, k] * s[atom(r), k]) * Win[k, g]
// rows r = flattened (atom, i, j); 16-row tile per wave; N = 32 (two 16-tiles).
// ---------------------------------------------------------------------------
__global__ void linear_in_kernel(const float* __restrict__ f,
                                 const float* __restrict__ s,
                                 const float* __restrict__ W,  // 32x32
                                 float* __restrict__ h, int Mrows, int sq) {
  int wave = blockIdx.x * (blockDim.x >> 5) + (threadIdx.x >> 5);
  int m0 = wave * 16;
  if (m0 >= Mrows) return;  // wave-uniform
  int lane = threadIdx.x & 31;
  int half = lane >> 4, lid = lane & 15;

  v8f acc0 = {}, acc1 = {};
  int arow = m0 + lid;
  int natom = arow / sq;
  const float* fr = f + (size_t)arow * 32;
  const float* sr = s + (size_t)natom * 32;
#pragma unroll
  for (int ks = 0; ks < 8; ++ks) {
    int k = ks * 4 + 2 * half;
    v2f a;
    a.x = fr[k] * sr[k];
    a.y = fr[k + 1] * sr[k + 1];
    v2f b0, b1;
    b0.x = W[k * 32 + lid];
    b0.y = W[(k + 1) * 32 + lid];
    b1.x = W[k * 32 + 16 + lid];
    b1.y = W[(k + 1) * 32 + 16 + lid];
    acc0 = wmma_f32_16x16x4(a, b0, acc0);
    acc1 = wmma_f32_16x16x4(a, b1, acc1);
  }
#pragma unroll
  for (int v = 0; v < 8; ++v) {
    size_t mr = (size_t)(m0 + v + 8 * half) * 32;
    h[mr + lid] = acc0[v];
    h[mr + 16 + lid] = acc1[v];
  }
}

// ---------------------------------------------------------------------------
// Kernel 3: radial MLP  rbm = silu(rb @ A) @ B
// One wave per 16 edges. Stage 1: 16x128 hidden in 8 WMMA accumulators,
// SiLU in registers, stash to padded LDS (stride 132 -> no bank conflicts).
// Stage 2: 16 x K_l output via WMMA with A-operand re-read from LDS.
// 4 waves/block -> 33,792 B static LDS.
// ---------------------------------------------------------------------------
__global__ void radial_mlp_kernel(const float* __restrict__ rb_l,
                                  const float* __restrict__ A_l,  // 32x128
                                  const float* __restrict__ B_l,  // 128xK_l
                                  float* __restrict__ rbm_l, int E, int Kl) {
  __shared__ float Hs[4 * 16 * 132];
  int wl = threadIdx.x >> 5;
  int wave = blockIdx.x * (blockDim.x >> 5) + wl;
  float* H = Hs + wl * 16 * 132;
  int e0 = wave * 16;
  int lane = threadIdx.x & 31;
  int half = lane >> 4, lid = lane & 15;

  if (e0 < E) {
    v8f acc[8] = {};
    const float* xr = rb_l + (size_t)(e0 + lid) * 32;
#pragma unroll
    for (int ks = 0; ks < 8; ++ks) {
      int k = ks * 4 + 2 * half;
      v2f a;
      a.x = xr[k];
      a.y = xr[k + 1];
#pragma unroll
      for (int nt = 0; nt < 8; ++nt) {
        v2f b;
        b.x = A_l[k * 128 + nt * 16 + lid];
        b.y = A_l[(k + 1) * 128 + nt * 16 + lid];
        acc[nt] = wmma_f32_16x16x4(a, b, acc[nt]);
      }
    }
    // SiLU + transpose-stash to LDS: H[row(edge)][hidden k]
#pragma unroll
    for (int nt = 0; nt < 8; ++nt)
#pragma unroll
      for (int v = 0; v < 8; ++v) {
        float x = acc[nt][v];
        H[(v + 8 * half) * 132 + nt * 16 + lid] = x / (1.0f + __expf(-x));
      }
  }
  __syncthreads();
  if (e0 >= E) return;

  int ntiles = Kl >> 4;  // 8/6/4/2
  v8f acc2[8] = {};
  for (int ks = 0; ks < 32; ++ks) {
    int k = ks * 4 + 2 * half;
    v2f a;
    a.x = H[lid * 132 + k];
    a.y = H[lid * 132 + k + 1];
    for (int nt = 0; nt < ntiles; ++nt) {
      v2f b;
      b.x = B_l[k * Kl + nt * 16 + lid];
      b.y = B_l[(k + 1) * Kl + nt * 16 + lid];
      acc2[nt] = wmma_f32_16x16x4(a, b, acc2[nt]);
    }
  }
  for (int nt = 0; nt < ntiles; ++nt)
#pragma unroll
    for (int v = 0; v < 8; ++v)
      rbm_l[(size_t)(e0 + v + 8 * half) * Kl + nt * 16 + lid] = acc2[nt][v];
}

// ---------------------------------------------------------------------------
// Kernel 4 (per l): vexp slices -> CG uncouple (U in LDS) -> gather h[nb] ->
// (P+1)x(P+1) per-channel TP -> atomic scatter-add into pooled[center].
// One wave per edge; lane = channel (32 channels per l-group) -> coalesced.
// ---------------------------------------------------------------------------
template <int L>
__global__ void edge_kernel(const float* __restrict__ sh0,
                            const float* __restrict__ sh1,
                            const float* __restrict__ sh2,
                            const float* __restrict__ sh3,
                            const float* __restrict__ rbm0,
                            const float* __restrict__ rbm1,
                            const float* __restrict__ rbm2,
                            const float* __restrict__ rbm3,
                            const float* __restrict__ U,
                            const float* __restrict__ h_l,
                            const int* __restrict__ centers,
                            const int* __restrict__ neighbors,
                            float* __restrict__ pooled_l, int E) {
  constexpr int P1 = (L <= 1) ? 2 : (L == 2 ? 3 : 4);
  constexpr int SQL = P1 * P1;
  constexpr int LOW = (L == 0) ? 96 : (L == 1) ? 64 : (L == 2) ? 32 : 0;
  __shared__ float Us[SQL * SQL];
  for (int t = threadIdx.x; t < SQL * SQL; t += blockDim.x) Us[t] = U[t];
  __syncthreads();

  int e = blockIdx.x * (blockDim.x >> 5) + (threadIdx.x >> 5);
  if (e >= E) return;
  int c = threadIdx.x & 31;

  // stacked[q] = concat_lp sh_lp[m] * rbm_lp[e, LOW + c], zero-padded
  float stacked[SQL];
  int q = 0;
  {
    float r0 = rbm0[(size_t)e * 128 + LOW + c];
    stacked[q++] = sh0[e] * r0;
  }
  if (L >= 1) {
    float r1 = rbm1[(size_t)e * 96 + LOW + c];
#pragma unroll
    for (int m = 0; m < 3; ++m) stacked[q++] = sh1[(size_t)e * 3 + m] * r1;
  }
  if (L >= 2) {
    float r2 = rbm2[(size_t)e * 64 + LOW + c];
#pragma unroll
    for (int m = 0; m < 5; ++m) stacked[q++] = sh2[(size_t)e * 5 + m] * r2;
  }
  if (L >= 3) {
    float r3 = rbm3[(size_t)e * 32 + LOW + c];
#pragma unroll
    for (int m = 0; m < 7; ++m) stacked[q++] = sh3[(size_t)e * 7 + m] * r3;
  }
  for (; q < SQL; ++q) stacked[q] = 0.0f;  // only L==0 pads

  // u = U @ stacked
  float u[SQL];
#pragma unroll
  for (int p = 0; p < SQL; ++p) {
    float a = 0.0f;
#pragma unroll
    for (int t = 0; t < SQL; ++t) a += Us[p * SQL + t] * stacked[t];
    u[p] = a;
  }

  // gather neighbor features for this channel
  int nb = neighbors[e];
  const float* hb = h_l + (size_t)nb * SQL * 32 + c;
  float hv[SQL];
#pragma unroll
  for (int t = 0; t < SQL; ++t) hv[t] = hb[(size_t)t * 32];

  // msg = (u as P1xP1) @ (hv as P1xP1) / sqrt(P1); scatter-add
  int ct = centers[e];
  float* pb = pooled_l + (size_t)ct * SQL * 32 + c;
  const float inv = (P1 == 2)   ? 0.70710678118654752f
                    : (P1 == 3) ? 0.57735026918962576f
                                : 0.5f;
#pragma unroll
  for (int i = 0; i < P1; ++i) {
#pragma unroll
    for (int k2 = 0; k2 < P1; ++k2) {
      float m = 0.0f;
#pragma unroll
      for (int j = 0; j < P1; ++j) m += u[i * P1 + j] * hv[j * P1 + k2];
      atomicAdd(&pb[(size_t)(i * P1 + k2) * 32], m * inv);
    }
  }
}

// ---------------------------------------------------------------------------
// Kernel 5: out[r, g] = f[r, g] + msg_scale * sum_k pooled[r, k] * Wout[k, g]
// ---------------------------------------------------------------------------
__global__ void linear_out_kernel(const float* __restrict__ pooled,
                                  const float* __restrict__ fin,
                                  const float* __restrict__ W,  // 32x32
                                  const float* __restrict__ msc,
                                  float* __restrict__ out, int Mrows) {
  int wave = blockIdx.x * (blockDim.x >> 5) + (threadIdx.x >> 5);
  int m0 = wave * 16;
  if (m0 >= Mrows) return;  // wave-uniform
  int lane = threadIdx.x & 31;
  int half = lane >> 4, lid = lane & 15;

  v8f acc0 = {}, acc1 = {};
  const float* pr = pooled + (size_t)(m0 + lid) * 32;
#pragma unroll
  for (int ks = 0; ks < 8; ++ks) {
    int k = ks * 4 + 2 * half;
    v2f a;
    a.x = pr[k];
    a.y = pr[k + 1];
    v2f b0, b1;
    b0.x = W[k * 32 + lid];
    b0.y = W[(k + 1) * 32 + lid];
    b1.x = W[k * 32 + 16 + lid];
    b1.y = W[(k + 1) * 32 + 16 + lid];
    acc0 = wmma_f32_16x16x4(a, b0, acc0);
    acc1 = wmma_f32_16x16x4(a, b1, acc1);
  }
  float ms = msc[0];
#pragma unroll
  for (int v = 0; v < 8; ++v) {
    size_t mr = (size_t)(m0 + v + 8 * half) * 32;
    out[mr + lid] = fin[mr + lid] + ms * acc0[v];
    out[mr + 16 + lid] = fin[mr + 16 + lid] + ms * acc1[v];
  }
}

// ---------------------------------------------------------------------------
// Host launcher
// ---------------------------------------------------------------------------
extern "C" void kernel_launch(void* const* d_in, const int* in_sizes, int n_in,
                              void* d_out, int out_size, void* d_ws,
                              size_t ws_size, hipStream_t stream) {
  (void)in_sizes; (void)n_in; (void)out_size; (void)ws_size;

  static const int SQh[4] = {4, 4, 9, 16};
  static const int KMAXh[4] = {128, 96, 64, 32};
  const int NA = 20000, NE = 80000;

  const float* rb = (const float*)d_in[0];
  const float* sh[4] = {(const float*)d_in[1], (const float*)d_in[2],
                        (const float*)d_in[3], (const float*)d_in[4]};
  const float* fin[4] = {(const float*)d_in[5], (const float*)d_in[6],
                         (const float*)d_in[7], (const float*)d_in[8]};
  const float* U1 = (const float*)d_in[9];
  const float* U2 = (const float*)d_in[10];
  const float* U3 = (const float*)d_in[11];
  const float* gammas = (const float*)d_in[12];
  const float* Win = (const float*)d_in[13];
  const float* Wout = (const float*)d_in[14];
  const float* A = (const float*)d_in[15];
  const float* B[4] = {(const float*)d_in[16], (const float*)d_in[17],
                       (const float*)d_in[18], (const float*)d_in[19]};
  const float* msc = (const float*)d_in[20];
  const int* centers = (const int*)d_in[21];
  const int* neighbors = (const int*)d_in[22];
  float* out = (float*)d_out;

  // workspace layout (floats)
  float* ws = (float*)d_ws;
  size_t off = 0;
  float* sL[4];
  for (int l = 0; l < 4; ++l) { sL[l] = ws + off; off += (size_t)NA * 32; }
  float* hL[4];
  for (int l = 0; l < 4; ++l) { hL[l] = ws + off; off += (size_t)NA * SQh[l] * 32; }
  float* rbmL[4];
  for (int l = 0; l < 4; ++l) { rbmL[l] = ws + off; off += (size_t)NE * KMAXh[l]; }
  size_t poolBase = off;
  float* poolL[4];
  for (int l = 0; l < 4; ++l) { poolL[l] = ws + off; off += (size_t)NA * SQh[l] * 32; }
  size_t poolN = off - poolBase;

  // 0) zero pooled accumulators (harness does not re-zero between replays)
  zero_kernel<<<(unsigned)((poolN + 255) / 256), 256, 0, stream>>>(ws + poolBase, poolN);

  // 1) RMS scales
  for (int l = 0; l < 4; ++l)
    rms_scale_kernel<<<(NA * 32 + 255) / 256, 256, 0, stream>>>(
        fin[l], gammas + l * 32, sL[l], NA, SQh[l]);

  // 2) linear_in (WMMA)
  for (int l = 0; l < 4; ++l) {
    int M = NA * SQh[l];
    linear_in_kernel<<<(M + 127) / 128, 256, 0, stream>>>(
        fin[l], sL[l], Win + l * 32 * 32, hL[l], M, SQh[l]);
  }

  // 3) radial MLP (WMMA, 4 waves/block)
  for (int l = 0; l < 4; ++l) {
    int tiles = NE / 16;
    radial_mlp_kernel<<<(tiles + 3) / 4, 128, 0, stream>>>(
        rb + (size_t)l * NE * 32, A + l * 32 * 128, B[l], rbmL[l], NE, KMAXh[l]);
  }

  // 4) edge kernels: uncouple + gather + TP + atomic scatter
  int eb = (NE + 7) / 8;  // 8 edges (waves) per 256-thread block
  edge_kernel<0><<<eb, 256, 0, stream>>>(sh[0], sh[1], sh[2], sh[3], rbmL[0],
                                         rbmL[1], rbmL[2], rbmL[3], U1, hL[0],
                                         centers, neighbors, poolL[0], NE);
  edge_kernel<1><<<eb, 256, 0, stream>>>(sh[0], sh[1], sh[2], sh[3], rbmL[0],
                                         rbmL[1], rbmL[2], rbmL[3], U1, hL[1],
                                         centers, neighbors, poolL[1], NE);
  edge_kernel<2><<<eb, 256, 0, stream>>>(sh[0], sh[1], sh[2], sh[3], rbmL[0],
                                         rbmL[1], rbmL[2], rbmL[3], U2, hL[2],
                                         centers, neighbors, poolL[2], NE);
  edge_kernel<3><<<eb, 256, 0, stream>>>(sh[0], sh[1], sh[2], sh[3], rbmL[0],
                                         rbmL[1], rbmL[2], rbmL[3], U3, hL[3],
                                         centers, neighbors, poolL[3], NE);

  // 5) linear_out + residual (WMMA) into d_out (concatenated per-l)
  size_t outOff[4] = {0, (size_t)NA * 4 * 32, (size_t)NA * 8 * 32,
                      (size_t)NA * 17 * 32};
  for (int l = 0; l < 4; ++l) {
    int M = NA * SQh[l];
    linear_out_kernel<<<(M + 127) / 128, 256, 0, stream>>>(
        poolL[l], fin[l], Wout + l * 32 * 32, msc, out + outOff[l], M);
  }
}